// BlockReLU_V1_51548197487422
// MI455X (gfx1250) — compile-verified
//
#include <hip/hip_runtime.h>

// BlockReLU for (N,C,H,W) = (32,128,112,112) fp32.
//   c[  0, 64): 1x1 blocks  -> plain ReLU (mask = x>=0)
//   c[ 64, 96): 2x2 blocks  -> mask = (block sum >= 0), broadcast over block
//   c[ 96,120): 4x4 blocks  -> mask = (block sum >= 0), broadcast over block
//   c[120,128): passthrough copy
// Pure one-pass streaming workload: ~411 MB traffic -> ~17.6 us floor @ 23.3 TB/s.
// Working set (205 MB in + 205 MB out, each touched once) exceeds the 192 MB L2,
// so all accesses use non-temporal hints (TH=NT) to avoid L2 churn.
// 128-bit coalesced loads/stores, per-thread in-register block sums, single
// fused kernel, wave32 blocks of 256 threads (8 waves/WGP-dispatch).

typedef float v4f __attribute__((ext_vector_type(4)));

namespace br {

constexpr int Nn  = 32;
constexpr int Cc  = 128;
constexpr int Hh  = 112;
constexpr int Ww  = 112;
constexpr int HW   = Hh * Ww;        // 12544
constexpr int CHW  = Cc * HW;        // 1,605,632
constexpr int HW4  = HW / 4;         // 3136  (float4 per channel-plane)
constexpr int CHW4 = CHW / 4;        // 401,408

constexpr int TPB = 256;

// Region A: relu, channels [0,64), one v4f/thread
constexpr int A_PER_N   = 64 * HW4;                       // 200,704 vec4
constexpr int BLK_A     = (A_PER_N * Nn) / TPB;           // 25,088
// Region B: 2x2, channels [64,96), thread = (c,hb,wq): 2 v4f (rows h,h+1)
constexpr int B_PER_N   = 32 * (Hh / 2) * (Ww / 4);       // 50,176 threads
constexpr int BLK_B     = (B_PER_N * Nn) / TPB;           // 6,272
// Region C: 4x4, channels [96,120), thread = one 4x4 block: 4 v4f
constexpr int C_PER_N   = 24 * (Hh / 4) * (Ww / 4);       // 18,816 threads
constexpr int BLK_C     = (C_PER_N * Nn) / TPB;           // 2,352
// Region D: copy, channels [120,128), one v4f/thread
constexpr int D_PER_N   = 8 * HW4;                        // 25,088 vec4
constexpr int BLK_D     = (D_PER_N * Nn) / TPB;           // 3,136

constexpr int A_END = BLK_A;                              // 25,088
constexpr int B_END = A_END + BLK_B;                      // 31,360
constexpr int C_END = B_END + BLK_C;                      // 33,712
constexpr int D_END = C_END + BLK_D;                      // 36,848

__device__ __forceinline__ float signmask(float s) { return (s >= 0.0f) ? 1.0f : 0.0f; }

// Non-temporal 128-bit load/store: one-pass data, bypass/minimize L2 residency.
__device__ __forceinline__ v4f ldnt(const v4f* p) { return __builtin_nontemporal_load(p); }
__device__ __forceinline__ void stnt(v4f* p, v4f v) { __builtin_nontemporal_store(v, p); }

} // namespace br

__global__ __launch_bounds__(br::TPB)
void blockrelu_fused_kernel(const float* __restrict__ in, float* __restrict__ out) {
    using namespace br;
    const int b = blockIdx.x;

    if (b < A_END) {
        // ---- Region A: plain ReLU on channels [0,64) ----
        const int tid = b * TPB + (int)threadIdx.x;          // vec4 id in region
        const int n   = tid / A_PER_N;
        const int rel = tid - n * A_PER_N;
        const int off4 = n * CHW4 + rel;                     // c0 = 0
        // Streaming prefetch hint (speculative; safe past end) -> global_prefetch_b8
        __builtin_prefetch((const void*)(((const v4f*)in) + off4 + 1024), 0, 0);
        v4f v = ldnt(((const v4f*)in) + off4);
        v.x *= signmask(v.x);
        v.y *= signmask(v.y);
        v.z *= signmask(v.z);
        v.w *= signmask(v.w);
        stnt(((v4f*)out) + off4, v);
    } else if (b < B_END) {
        // ---- Region B: 2x2 blocks on channels [64,96) ----
        // Each thread: 4 columns x 2 rows = two 2x2 blocks, two v4f loads.
        int tid = (b - A_END) * TPB + (int)threadIdx.x;
        const int wq = tid % (Ww / 4);  tid /= (Ww / 4);     // 0..27
        const int hb = tid % (Hh / 2);  tid /= (Hh / 2);     // 0..55
        const int c  = tid % 32;                             // 0..31
        const int n  = tid / 32;                             // 0..31
        const int base = n * CHW + (64 + c) * HW + (2 * hb) * Ww + 4 * wq;
        v4f r0 = ldnt((const v4f*)(in + base));
        v4f r1 = ldnt((const v4f*)(in + base + Ww));
        const float m0 = signmask((r0.x + r0.y) + (r1.x + r1.y));
        const float m1 = signmask((r0.z + r0.w) + (r1.z + r1.w));
        r0.x *= m0; r0.y *= m0; r1.x *= m0; r1.y *= m0;
        r0.z *= m1; r0.w *= m1; r1.z *= m1; r1.w *= m1;
        stnt((v4f*)(out + base),      r0);
        stnt((v4f*)(out + base + Ww), r1);
    } else if (b < C_END) {
        // ---- Region C: 4x4 blocks on channels [96,120) ----
        // Each thread owns one full 4x4 block: 4 v4f loads, 16-elem sum in regs.
        int tid = (b - B_END) * TPB + (int)threadIdx.x;
        const int wq = tid % (Ww / 4);  tid /= (Ww / 4);     // 0..27
        const int hb = tid % (Hh / 4);  tid /= (Hh / 4);     // 0..27
        const int c  = tid % 24;                             // 0..23
        const int n  = tid / 24;                             // 0..31
        const int base = n * CHW + (96 + c) * HW + (4 * hb) * Ww + 4 * wq;
        __builtin_prefetch((const void*)(in + base + 8 * Ww), 0, 0);
        v4f r0 = ldnt((const v4f*)(in + base + 0 * Ww));
        v4f r1 = ldnt((const v4f*)(in + base + 1 * Ww));
        v4f r2 = ldnt((const v4f*)(in + base + 2 * Ww));
        v4f r3 = ldnt((const v4f*)(in + base + 3 * Ww));
        const float s = (((r0.x + r0.y) + (r0.z + r0.w)) +
                         ((r1.x + r1.y) + (r1.z + r1.w))) +
                        (((r2.x + r2.y) + (r2.z + r2.w)) +
                         ((r3.x + r3.y) + (r3.z + r3.w)));
        const float m = signmask(s);
        r0.x *= m; r0.y *= m; r0.z *= m; r0.w *= m;
        r1.x *= m; r1.y *= m; r1.z *= m; r1.w *= m;
        r2.x *= m; r2.y *= m; r2.z *= m; r2.w *= m;
        r3.x *= m; r3.y *= m; r3.z *= m; r3.w *= m;
        stnt((v4f*)(out + base + 0 * Ww), r0);
        stnt((v4f*)(out + base + 1 * Ww), r1);
        stnt((v4f*)(out + base + 2 * Ww), r2);
        stnt((v4f*)(out + base + 3 * Ww), r3);
    } else {
        // ---- Region D: passthrough copy on channels [120,128) ----
        const int tid = (b - C_END) * TPB + (int)threadIdx.x;
        const int n   = tid / D_PER_N;
        const int rel = tid - n * D_PER_N;
        const int off4 = n * CHW4 + 120 * HW4 + rel;
        stnt(((v4f*)out) + off4, ldnt(((const v4f*)in) + off4));
    }
}

extern "C" void kernel_launch(void* const* d_in, const int* in_sizes, int n_in,
                              void* d_out, int out_size, void* d_ws, size_t ws_size,
                              hipStream_t stream) {
    (void)in_sizes; (void)n_in; (void)out_size; (void)d_ws; (void)ws_size;
    const float* act = (const float*)d_in[0];
    float* out = (float*)d_out;
    dim3 grid(br::D_END);   // 36,848 blocks: A|B|C|D regions, all exact multiples of 256
    dim3 block(br::TPB);
    blockrelu_fused_kernel<<<grid, block, 0, stream>>>(act, out);
}